// LSNet_75737453298097
// MI455X (gfx1250) — compile-verified
//
#include <hip/hip_runtime.h>
#include <hip/hip_bf16.h>

typedef float v2f __attribute__((ext_vector_type(2)));
typedef float v8f __attribute__((ext_vector_type(8)));

#define VOL    (128 * 128 * 128)
#define NV     (2 * VOL)          // total elements of inp / out
#define NPART  1024
#define CHUNK  (NV / NPART)       // 4096

// ---------------- Stage 1: sum(inp) -> partial sums (fixed-order, deterministic)
__global__ __launch_bounds__(256) void reduce_partial_kernel(
    const float* __restrict__ inp, float* __restrict__ part) {
  __shared__ float red[256];
  const int b = blockIdx.x, t = threadIdx.x;
  const float* p = inp + (size_t)b * CHUNK;
  float s = 0.f;
  for (int i = t; i < CHUNK; i += 256) s += p[i];
  red[t] = s;
  __syncthreads();
  for (int off = 128; off > 0; off >>= 1) {
    if (t < off) red[t] += red[t + off];
    __syncthreads();
  }
  if (t == 0) part[b] = red[0];
}

// ---------------- Stage 2: finalize c2; store (c2^2, -2*c2)
__global__ __launch_bounds__(256) void reduce_final_kernel(
    const float* __restrict__ part, float* __restrict__ cons) {
  __shared__ float red[256];
  const int t = threadIdx.x;
  float s = 0.f;
  for (int i = t; i < NPART; i += 256) s += part[i];
  red[t] = s;
  __syncthreads();
  for (int off = 128; off > 0; off >>= 1) {
    if (t < off) red[t] += red[t + off];
    __syncthreads();
  }
  if (t == 0) {
    float c2 = red[0] / ((float)NV + 1e-8f);
    cons[0] = c2 * c2;   // c2^2
    cons[1] = -2.f * c2; // -2*c2
  }
}

// ---------------- Stage 3: out = tanh(conv3x3x3(c2^2 - 2*c2*inp, c1_w)) via WMMA
// Each workgroup: one (n, d, 16-wide w tile), full H=128.
// Each wave (wave32): one 16x16 (h x w) output tile.
// kw-axis conv as banded matmul: 9 (kd,kh) planes x 5 K-chunks of
// v_wmma_f32_16x16x4_f32 accumulating into one 16x16 fp32 tile.
__global__ __launch_bounds__(256) void conv_tanh_wmma_kernel(
    const float* __restrict__ inp, const float* __restrict__ w27,
    const float* __restrict__ cons, float* __restrict__ out) {
  __shared__ float S[9][20][16];   // banded weight matrices, K padded 18->20
  __shared__ float T[3][130][20];  // 3 d-slices, 130 h rows, 18+2 w cols (zero halo)

  const int tid = threadIdx.x;
  const int wt  = blockIdx.x;   // w tile: 0..7
  const int d   = blockIdx.y;   // 0..127
  const int n   = blockIdx.z;   // 0..1
  const int w0  = wt * 16;
  const float c2sq = cons[0];
  const float m2c2 = cons[1];

  // Build banded weight matrices: S[p][c][col] = w27[p*3 + (c-col)] if c-col in [0,2]
  for (int idx = tid; idx < 9 * 20 * 16; idx += 256) {
    int p = idx / 320, rem = idx % 320, c = rem / 16, col = rem % 16;
    int kw = c - col;
    float v = 0.f;
    if (c < 18 && kw >= 0 && kw <= 2) v = w27[p * 3 + kw];
    S[p][c][col] = v;
  }

  // Stage data tile: q = c2^2 - 2*c2*inp, with zero padding (matches conv zero-pad,
  // since the padded difference (0-c2)^2-0^2 never exists: both convs pad the
  // *squared* tensors with literal zeros, so padded taps contribute 0).
  const size_t nbase = (size_t)n * VOL;
  for (int idx = tid; idx < 3 * 130 * 20; idx += 256) {
    int kd = idx / (130 * 20), rem = idx % (130 * 20), r = rem / 20, c = rem % 20;
    int dz = d + kd - 1, hy = r - 1, wx = w0 - 1 + c;
    float v = 0.f;
    if (c < 18 && (unsigned)dz < 128u && (unsigned)hy < 128u && (unsigned)wx < 128u) {
      float t = inp[nbase + ((size_t)dz * 128 + hy) * 128 + wx];
      v = fmaf(m2c2, t, c2sq);
    }
    T[kd][r][c] = v;
  }
  __syncthreads();

  const int lane = tid & 31;
  const int wid  = tid >> 5;     // 8 waves -> 8 h-tiles of 16
  const int m    = lane & 15;    // A: row M / B: col N
  const int g    = lane >> 4;    // half-wave selects K pair
  const int h0   = wid * 16;

  v8f acc = {0.f, 0.f, 0.f, 0.f, 0.f, 0.f, 0.f, 0.f};
  for (int kd = 0; kd < 3; ++kd) {
    for (int kh = 0; kh < 3; ++kh) {
      const int p = kd * 3 + kh;
      const float* trow = &T[kd][h0 + kh + m][0];  // 16-row strip, conflict-free (stride 20)
#pragma unroll
      for (int kc = 0; kc < 5; ++kc) {
        const int c0 = kc * 4 + g * 2;
        v2f a, b;
        // A 16x4 fp32 layout: lane m+16g holds A[m, 2g] (v0), A[m, 2g+1] (v1)
        a.x = trow[c0];
        a.y = trow[c0 + 1];
        // B 4x16 fp32 layout (mirror): lane n+16g holds B[2g, n], B[2g+1, n]
        b.x = S[p][c0][m];
        b.y = S[p][c0 + 1][m];
        acc = __builtin_amdgcn_wmma_f32_16x16x4_f32(
            /*neg_a=*/false, a, /*neg_b=*/false, b,
            /*c_mod=*/(short)0, acc, /*reuse_a=*/false, /*reuse_b=*/false);
      }
    }
  }

  // D 16x16 fp32: lanes 0-15 hold M=r (VGPR r), N=lane; lanes 16-31 hold M=8+r, N=lane-16
  const int mbase = (lane < 16) ? 0 : 8;
  float* ocol = out + nbase + (size_t)d * 128 * 128 + w0 + m;
#pragma unroll
  for (int r = 0; r < 8; ++r) {
    int h = h0 + mbase + r;
    ocol[(size_t)h * 128] = tanhf(acc[r]);
  }
}

extern "C" void kernel_launch(void* const* d_in, const int* in_sizes, int n_in,
                              void* d_out, int out_size, void* d_ws, size_t ws_size,
                              hipStream_t stream) {
  // setup_inputs order: x, inp, conv_w, conv_b, bn_g, bn_b, c1_w, c1_b
  // Dead-code analysis: output depends only on inp and c1_w (c1_b cancels).
  const float* inp = (const float*)d_in[1];
  const float* c1w = (const float*)d_in[6];
  float* out  = (float*)d_out;
  float* part = (float*)d_ws;        // NPART partial sums
  float* cons = part + NPART;        // {c2^2, -2*c2}

  reduce_partial_kernel<<<NPART, 256, 0, stream>>>(inp, part);
  reduce_final_kernel<<<1, 256, 0, stream>>>(part, cons);
  conv_tanh_wmma_kernel<<<dim3(8, 128, 2), 256, 0, stream>>>(inp, c1w, cons, out);
}